// Model_76622216561223
// MI455X (gfx1250) — compile-verified
//
#include <hip/hip_runtime.h>
#include <hip/hip_bf16.h>

// ---------------------------------------------------------------------------
// PointNet++ sample-and-group for MI455X (gfx1250, wave32).
//   B=8, N=8192, C=128, GROUPS=2048, K=24, RADIUS=0.2
// Kernel 1: persistent FPS, one workgroup per batch (serial 2048-step argmax).
// Kernel 2: ball query via V_WMMA_F32_16X16X4_F32 distance tiles
//           (A=[-2x,-2y,-2z,|a|^2], B=[x,y,z,1]^T embedding), 2x-unrolled
//           WMMA pipeline, ballot-based ordered index selection, then fused
//           gather/concat writeout with non-temporal (L2-bypassing) stores.
// ---------------------------------------------------------------------------

#define NB     8
#define NPTS   8192
#define NCH    128
#define NGRP   2048
#define NK     24
#define R2     0.04f

typedef __attribute__((ext_vector_type(2))) float v2f;
typedef __attribute__((ext_vector_type(8))) float v8f;

__device__ __forceinline__ unsigned ballot32(bool p) {
    return __builtin_amdgcn_ballot_w32(p);
}

// ---------------------------------------------------------------------------
// Kernel 1: farthest point sampling. grid=(B), block=(1024)=32 waves.
// Each thread owns 8 points (coords + running min-distance in VGPRs).
// LDS: xyz of whole batch (for centroid lookup) + 32-entry reduction stage.
// ---------------------------------------------------------------------------
__global__ void fps_kernel(const float* __restrict__ xyz,
                           float* __restrict__ new_xyz,   // (B, NGRP, 3)
                           int* __restrict__ fps_idx) {   // (B, NGRP)
    extern __shared__ float fsm[];
    float* px = fsm;                 // NPTS
    float* py = fsm + NPTS;          // NPTS
    float* pz = fsm + 2 * NPTS;      // NPTS
    float* wval = fsm + 3 * NPTS;    // 32
    int*   widx = (int*)(fsm + 3 * NPTS + 32);   // 32
    int*   sfar = widx + 32;                     // 1

    const int b   = blockIdx.x;
    const int tid = threadIdx.x;
    const int lane = tid & 31;
    const int wv   = tid >> 5;
    const float* bxyz = xyz + (size_t)b * NPTS * 3;

    for (int i = tid; i < NPTS; i += 1024) {
        px[i] = bxyz[i * 3 + 0];
        py[i] = bxyz[i * 3 + 1];
        pz[i] = bxyz[i * 3 + 2];
    }
    __syncthreads();

    // Private copies: 8 points per thread, strided.
    float rx[8], ry[8], rz[8], rd[8];
#pragma unroll
    for (int k = 0; k < 8; ++k) {
        int i = tid + k * 1024;
        rx[k] = px[i]; ry[k] = py[i]; rz[k] = pz[i];
        rd[k] = 1.0e10f;
    }

    int farthest = 0;
    for (int it = 0; it < NGRP; ++it) {
        if (tid == 0) {
            fps_idx[b * NGRP + it] = farthest;
            float* o = new_xyz + ((size_t)b * NGRP + it) * 3;
            o[0] = px[farthest]; o[1] = py[farthest]; o[2] = pz[farthest];
        }
        const float cx = px[farthest], cy = py[farthest], cz = pz[farthest];

        float bv = -1.0f; int bi = NPTS;
#pragma unroll
        for (int k = 0; k < 8; ++k) {
            float dx = rx[k] - cx, dy = ry[k] - cy, dz = rz[k] - cz;
            float d = dx * dx + dy * dy + dz * dz;
            rd[k] = fminf(rd[k], d);
            int i = tid + k * 1024;
            if (rd[k] > bv || (rd[k] == bv && i < bi)) { bv = rd[k]; bi = i; }
        }
        // wave32 argmax reduce (max value, ties -> smallest index == argmax)
#pragma unroll
        for (int off = 16; off > 0; off >>= 1) {
            float ov = __shfl_xor(bv, off, 32);
            int   oi = __shfl_xor(bi, off, 32);
            if (ov > bv || (ov == bv && oi < bi)) { bv = ov; bi = oi; }
        }
        if (lane == 0) { wval[wv] = bv; widx[wv] = bi; }
        __syncthreads();
        if (wv == 0) {
            float v = wval[lane]; int i = widx[lane];
#pragma unroll
            for (int off = 16; off > 0; off >>= 1) {
                float ov = __shfl_xor(v, off, 32);
                int   oi = __shfl_xor(i, off, 32);
                if (ov > v || (ov == v && oi < i)) { v = ov; i = oi; }
            }
            if (lane == 0) sfar[0] = i;
        }
        __syncthreads();
        farthest = sfar[0];
    }
}

// ---------------------------------------------------------------------------
// Kernel 2: ball query with WMMA distance tiles + fused gather/concat.
// grid=(NGRP/128, B), block=(256)=8 waves; each wave owns 16 anchors.
// Per iteration: TWO independent V_WMMA_F32_16X16X4_F32 (32 points), each
//   D[m][n] = -2*dot(anchor_m, pt_n) + |anchor_m|^2,   dist = D + |pt_n|^2.
// 16 ballots -> per-row 32-bit qualify mask (bit p <-> index base+p, order
// preserved) -> ordered index lists, early exit at 24 hits per row.
// ---------------------------------------------------------------------------
__global__ void group_kernel(const float* __restrict__ xyz,
                             const float* __restrict__ points,
                             const float* __restrict__ new_xyz,
                             const int* __restrict__ fps_idx,
                             float* __restrict__ new_points) { // (B,NGRP,NK,259)
    extern __shared__ float smem[];
    float* sx = smem;                // NPTS
    float* sy = smem + NPTS;         // NPTS
    float* sz = smem + 2 * NPTS;     // NPTS
    int*   slist = (int*)(smem + 3 * NPTS);  // 8 waves * 16 rows * NK

    const int b    = blockIdx.y;
    const int tid  = threadIdx.x;
    const int lane = tid & 31;
    const int wv   = tid >> 5;
    const int g0   = (blockIdx.x * 8 + wv) * 16;   // first anchor of this wave
    const float* bxyz = xyz + (size_t)b * NPTS * 3;
    const float* pts  = points + (size_t)b * NPTS * NCH;

    for (int i = tid; i < NPTS; i += 256) {
        sx[i] = bxyz[i * 3 + 0];
        sy[i] = bxyz[i * 3 + 1];
        sz[i] = bxyz[i * 3 + 2];
    }
    __syncthreads();

    // ---- A tile (16 anchors x 4), embedded: row m = [-2x,-2y,-2z,|a|^2] ----
    const int m = lane & 15;
    const float* nx = new_xyz + ((size_t)b * NGRP + g0 + m) * 3;
    const float ax = nx[0], ay = nx[1], az = nx[2];
    const float an = fmaf(ax, ax, fmaf(ay, ay, az * az));
    v2f av;
    av.x = (lane < 16) ? (-2.0f * ax) : (-2.0f * az);
    av.y = (lane < 16) ? (-2.0f * ay) : an;

    int* mylist = &slist[(wv * 16 + m) * NK];
    unsigned cnt = 0;
    const unsigned sh = ((lane >> 3) & 1) * 16;

    for (int chunk = 0; chunk < NPTS / 16; chunk += 2) {
        const int base = chunk * 16;
        const int n0 = base + (lane & 15);
        const int n1 = n0 + 16;

        const float bx0 = sx[n0], by0 = sy[n0], bz0 = sz[n0];
        const float bx1 = sx[n1], by1 = sy[n1], bz1 = sz[n1];
        const float bn0 = fmaf(bx0, bx0, fmaf(by0, by0, bz0 * bz0));
        const float bn1 = fmaf(bx1, bx1, fmaf(by1, by1, bz1 * bz1));
        v2f bv0, bv1;                           // B col n = [x,y,z,1]^T
        bv0.x = (lane < 16) ? bx0 : bz0;
        bv0.y = (lane < 16) ? by0 : 1.0f;
        bv1.x = (lane < 16) ? bx1 : bz1;
        bv1.y = (lane < 16) ? by1 : 1.0f;

        v8f acc0 = {}, acc1 = {};
        acc0 = __builtin_amdgcn_wmma_f32_16x16x4_f32(
            false, av, false, bv0, (short)0, acc0, false, false);
        acc1 = __builtin_amdgcn_wmma_f32_16x16x4_f32(
            false, av, false, bv1, (short)0, acc1, false, false);

        // Row m's qualify bits: half of ballot #(m&7), chunks packed lo/hi.
        unsigned m32 = 0;
#pragma unroll
        for (int r = 0; r < 8; ++r) {
            unsigned fb0 = ballot32(acc0[r] + bn0 <= R2);
            unsigned fb1 = ballot32(acc1[r] + bn1 <= R2);
            if ((lane & 7) == r)
                m32 = ((fb0 >> sh) & 0xFFFFu) | (((fb1 >> sh) & 0xFFFFu) << 16);
        }
        if (lane < 16) {
            while (m32 && cnt < NK) {
                int p = __ffs(m32) - 1;        // ascending index order
                mylist[cnt] = base + p;
                ++cnt;
                m32 &= m32 - 1;
            }
        }
        if (ballot32(lane >= 16 || cnt >= NK) == 0xFFFFFFFFu) break;
    }

    // Fill remainder with first qualifying index (reference semantics).
    if (lane < 16) {
        int first = (cnt > 0) ? mylist[0] : fps_idx[b * NGRP + g0 + m];
        for (unsigned k = cnt; k < NK; ++k) mylist[k] = first;
    }

    // ---- fused gather + concat: [points[idx](128) | xyz[idx](3) | anchor(128)]
    // Output is write-once streaming (407 MB > L2): use non-temporal stores so
    // the hot `points` working set (32 MB) stays resident in the 192 MB L2.
    for (int mm = 0; mm < 16; ++mm) {
        const int g = g0 + mm;
        const int aidx = fps_idx[b * NGRP + g];
        const float* arow = pts + (size_t)aidx * NCH;
        float af[4];                      // anchor row, reused for all 24 k
#pragma unroll
        for (int c4 = 0; c4 < 4; ++c4) af[c4] = arow[lane + c4 * 32];

        const int* lst = &slist[(wv * 16 + mm) * NK];
        for (int k = 0; k < NK; ++k) {
            const int idx = lst[k];
            const float* prow = pts + (size_t)idx * NCH;
            float* orow = new_points + (((size_t)b * NGRP + g) * NK + k) * 259;

#pragma unroll
            for (int c4 = 0; c4 < 4; ++c4) {          // channels [0,128)
                int c = lane + c4 * 32;
                __builtin_nontemporal_store(prow[c], orow + c);
            }
            if (lane < 3) {                            // channels [128,131)
                float v = (lane == 0) ? sx[idx] : (lane == 1) ? sy[idx] : sz[idx];
                __builtin_nontemporal_store(v, orow + 128 + lane);
            }
#pragma unroll
            for (int c4 = 0; c4 < 4; ++c4) {          // channels [131,259)
                int c = lane + c4 * 32;
                __builtin_nontemporal_store(af[c4], orow + 131 + c);
            }
        }
    }
}

// ---------------------------------------------------------------------------
extern "C" void kernel_launch(void* const* d_in, const int* in_sizes, int n_in,
                              void* d_out, int out_size, void* d_ws, size_t ws_size,
                              hipStream_t stream) {
    const float* xyz    = (const float*)d_in[0];   // (8, 8192, 3)
    const float* points = (const float*)d_in[1];   // (8, 8192, 128)
    float* out = (float*)d_out;
    float* new_xyz    = out;                                 // (8,2048,3)
    float* new_points = out + (size_t)NB * NGRP * 3;         // (8,2048,24,259)
    int*   fps_idx    = (int*)d_ws;                          // (8,2048)

    const size_t fps_lds = (size_t)(3 * NPTS + 32) * 4 + 32 * 4 + 4;
    fps_kernel<<<dim3(NB), dim3(1024), fps_lds, stream>>>(xyz, new_xyz, fps_idx);

    const size_t grp_lds = (size_t)3 * NPTS * 4 + (size_t)8 * 16 * NK * 4;
    group_kernel<<<dim3(NGRP / 128, NB), dim3(256), grp_lds, stream>>>(
        xyz, points, new_xyz, fps_idx, new_points);
}